// ConvCaps_7078106104124
// MI455X (gfx1250) — compile-verified
//
#include <hip/hip_runtime.h>

// Problem constants (from reference): b=4, B=C=32, K=3, S=2, WIN=11, WOUT=5, ITER=3
#define NB    4
#define BC    32
#define CC    32
#define KK    3
#define WOUTD 5
#define WIND  11
#define PPD   15               // KK*WOUTD
#define XBSTR (544*121)        // per-batch stride of x (17*32 channels * 11*11)

typedef float v2f __attribute__((ext_vector_type(2)));
typedef float v8f __attribute__((ext_vector_type(8)));

// workspace segment sizes (floats)
#define N_VOTES (NB*BC*KK*KK*WOUTD*WOUTD*CC*16)   // 14,745,600
#define N_AWIN  (NB*BC*PPD*PPD)                   // 28,800
#define N_RW    (NB*BC*PPD*PPD*CC)                // 921,600
#define N_S0    (NB*WOUTD*WOUTD*CC)               // 3,200
#define N_S1    (N_S0*16)                         // 51,200
#define N_s1b   (BC*PPD*PPD)                      // 7,200

__device__ __constant__ int c_perm[3] = {0, 2, 1};
// duplicate-index tables for ii = [0,1,2,2,3,4,4,5,6,6,7,8,8,9,10]
__device__ __constant__ int c_dupA[15][2] = {
  {0,0},{1,1},{2,3},{2,3},{4,4},{5,6},{5,6},{7,7},{8,9},{8,9},{10,10},{11,12},{11,12},{13,13},{14,14}};
__device__ __constant__ int c_dupN[15] = {1,1,2,2,1,2,2,1,2,2,1,2,2,1,1};

// ---- K0: a_win gather + R window init (R only ever read through its window view) ----
__global__ void kInit(const float* __restrict__ x, float* __restrict__ awin,
                      float* __restrict__ Rw) {
  int idx = blockIdx.x * blockDim.x + threadIdx.x;
  if (idx < N_RW) Rw[idx] = 1.0f / 800.0f;                 // 1/(wout*wout*C)
  if (idx < N_AWIN) {
    int b2 = idx % PPD; int t = idx / PPD;
    int a  = t % PPD;   t /= PPD;
    int n  = t % BC;    int b = t / BC;
    int i = a / 3, k = a % 3, j = b2 / 3, l = b2 % 3;
    awin[idx] = x[b * XBSTR + (512 + n) * 121 + (2 * i + k) * 11 + (2 * j + l)];
  }
}

// ---- K1: votes via V_WMMA_F32_16X16X4_F32 ----
// One wave per 16x16 tile: M=(c4,p), N=(pos4,q), Kdim=r. 57,600 tiles total.
__global__ __launch_bounds__(128) void kVotes(const float* __restrict__ x,
                                              const float* __restrict__ W,
                                              float* __restrict__ votes) {
  int lane = threadIdx.x & 31;
  int tile = blockIdx.x * 4 + (threadIdx.x >> 5);
  int cg = tile % 8;  int t = tile / 8;     // capsule-C group of 4
  int pg = t % 25;    t /= 25;              // position group of 4 (b*5*5 = 100 pos)
  int n  = t % 32;    t /= 32;
  int l  = t % 3;     int k = t / 3;

  // A (16x4 f32): lanes 0-15 hold M=row with K=0(v0)/1(v1); lanes 16-31: K=2/3
  int row = lane & 15;
  int kb  = (lane >> 4) << 1;               // r base: 0 or 2
  int cA  = cg * 4 + (row >> 2);
  int p   = row & 3;
  const float* Wp = W + ((((k * 3 + l) * 32 + n) * 32 + cA) * 4 + p) * 4;
  v2f A; A.x = Wp[kb]; A.y = Wp[kb + 1];

  // B (4x16 f32): rows r striped across lanes (N=col), r=kb(v0)/kb+1(v1)
  int col = lane & 15;
  int pos = pg * 4 + (col >> 2);
  int q   = col & 3;
  int bb  = pos / 25; int rem = pos % 25;
  int xx  = rem / 5;  int yy = rem % 5;
  const float* xp = x + bb * XBSTR + (q * 32 + n) * 121 + (2 * xx + k) * 11 + (2 * yy + l);
  v2f Bv; Bv.x = xp[kb * 128 * 121]; Bv.y = xp[(kb + 1) * 128 * 121];

  v8f C = {};
  v8f D = __builtin_amdgcn_wmma_f32_16x16x4_f32(false, A, false, Bv,
                                                (short)0, C, false, false);

  // D: lanes 0-15 -> M=vi, lanes 16-31 -> M=vi+8; N=col
  int hf = lane >> 4;
  int base = ((((((bb * 32 + n) * 3 + k) * 3 + l) * 5 + xx) * 5 + yy)) * 32;
#pragma unroll
  for (int vi = 0; vi < 8; ++vi) {
    int Mrow = vi + hf * 8;
    int cD = cg * 4 + (Mrow >> 2);
    int m  = (Mrow & 3) * 4 + q;            // m = p*4+q
    votes[(base + cD) * 16 + m] = D[vi];
  }
}

// ---- K2: one sweep over votes -> S0, S1, S2 per (b,i,j,c[,m]) ----
__global__ void kSpass(const float* __restrict__ votes, const float* __restrict__ Rw,
                       const float* __restrict__ awin, float* __restrict__ S0,
                       float* __restrict__ S1, float* __restrict__ S2) {
  int tid = blockIdx.x * blockDim.x + threadIdx.x;
  if (tid >= N_S1) return;
  int m = tid & 15;
  int rest = tid >> 4;                      // (b*25+ij)*32 + c
  int c = rest & 31;
  int bij = rest >> 5;
  int ij = bij % 25; int b = bij / 25;
  int i = ij / 5, j = ij % 5;
  float s0 = 0.f, s1 = 0.f, s2 = 0.f;
  for (int n = 0; n < 32; ++n) {
    int bn = b * 32 + n;
    for (int k = 0; k < 3; ++k)
      for (int l = 0; l < 3; ++l) {
        int ridx = ((bn * PPD + (3 * i + k)) * PPD + (3 * j + l));
        float rh = Rw[ridx * 32 + c] * awin[ridx];
        float vv = votes[(((((bn * 3 + k) * 3 + l) * 5 + i) * 5 + j) * 32 + c) * 16 + m];
        s0 += rh; s1 += rh * vv; s2 += rh * vv * vv;
      }
  }
  S1[tid] = s1; S2[tid] = s2;
  if (m == 0) S0[rest] = s0;
}

// ---- K3: mu/sigma/acts/prodmu; writes acts & mu straight into d_out ----
__global__ void kStats(const float* __restrict__ S0, const float* __restrict__ S1,
                       const float* __restrict__ S2, const float* __restrict__ beta_v,
                       const float* __restrict__ lambda_, const float* __restrict__ beta_a,
                       float* __restrict__ mu, float* __restrict__ sigma,
                       float* __restrict__ acts, float* __restrict__ rspm,
                       float* __restrict__ dout) {
  int tid = blockIdx.x * blockDim.x + threadIdx.x;   // b*800 + ij*32 + c
  if (tid >= N_S0) return;
  float srh = S0[tid % 800];                // reference uses batch-0 sums only
  float s0b = S0[tid];
  float inv = 1.0f / srh;
  float bv = beta_v[0], lam = lambda_[0], ba = beta_a[0];
  float costsum = 0.f, prod = 1.f;
#pragma unroll
  for (int m = 0; m < 16; ++m) {
    float s1 = S1[tid * 16 + m], s2 = S2[tid * 16 + m];
    float muv = s1 * inv;
    float sg  = (s2 - 2.f * muv * s1 + muv * muv * s0b) * inv;  // Σr(v-mu)^2/srh
    mu[tid * 16 + m] = muv;
    sigma[tid * 16 + m] = sg;
    dout[3200 + tid * 16 + m] = muv;        // mus output
    costsum += bv + __logf(sg);
    prod *= 6.283185307179586f * muv;
  }
  float av = 1.0f / (1.0f + __expf(-lam * (ba - costsum * srh)));
  acts[tid] = av;
  dout[tid] = av;                           // acts output
  rspm[tid] = rsqrtf(prod);
}

// ---- K4: ap = acts * exp(se)/sqrt(prodmu); LDS-reduce over (b,c) -> s1[n,a,b2] ----
__global__ __launch_bounds__(128) void kAp(const float* __restrict__ votes,
                                           const float* __restrict__ sigma,
                                           const float* __restrict__ acts,
                                           const float* __restrict__ rspm,
                                           float* __restrict__ ap,
                                           float* __restrict__ s1out) {
  __shared__ float sm[128];
  int blk = blockIdx.x;                     // n*225 + a*15 + b2
  int n = blk / 225; int ab = blk % 225;
  int a = ab / 15, b2 = ab % 15;
  int i = a / 3, k = a % 3, j = b2 / 3, l = b2 % 3;
  int pk = c_perm[k], pl = c_perm[l];
  int t = threadIdx.x;
  int b = t >> 5, c = t & 31;
  int bijc = (b * 25 + i * 5 + j) * 32 + c;
  const float* vp = votes + (((((((b * 32 + n) * 3 + pk) * 3 + pl) * 5 + i) * 5 + j) * 32 + c)) * 16;
  const float* sg = sigma + bijc * 16;
  float se = 0.f;
#pragma unroll
  for (int m = 0; m < 16; ++m) { float d = vp[m] - sg[m]; se += d * d; }
  float apv = acts[bijc] * __expf(se) * rspm[bijc];
  ap[(((b * 32 + n) * PPD + a) * PPD + b2) * 32 + c] = apv;
  sm[t] = apv;
  __syncthreads();
  for (int s = 64; s > 0; s >>= 1) {
    if (t < s) sm[t] += sm[t + s];
    __syncthreads();
  }
  if (t == 0) s1out[blk] = sm[0];
}

// ---- K5: Rw = ap / Dg, Dg gathered deterministically via duplicate tables ----
__global__ void kUpdate(const float* __restrict__ ap, const float* __restrict__ s1,
                        float* __restrict__ Rw) {
  int idx = blockIdx.x * blockDim.x + threadIdx.x;
  if (idx >= N_RW) return;
  int c = idx & 31; int r = idx >> 5;
  int b2 = r % 15; r /= 15;
  int a = r % 15;  r /= 15;
  int n = r % 32;
  float Dg = 0.f;
  int na = c_dupN[a], nb = c_dupN[b2];
  for (int u = 0; u < na; ++u)
    for (int v = 0; v < nb; ++v)
      Dg += s1[n * 225 + c_dupA[a][u] * 15 + c_dupA[b2][v]];
  Rw[idx] = ap[idx] / Dg;
  (void)c;
}

extern "C" void kernel_launch(void* const* d_in, const int* in_sizes, int n_in,
                              void* d_out, int out_size, void* d_ws, size_t ws_size,
                              hipStream_t stream) {
  const float* x       = (const float*)d_in[0];
  const float* W       = (const float*)d_in[1];
  const float* beta_v  = (const float*)d_in[2];
  const float* lambda_ = (const float*)d_in[3];
  const float* beta_a  = (const float*)d_in[4];
  float* out = (float*)d_out;
  float* ws  = (float*)d_ws;
  (void)in_sizes; (void)n_in; (void)out_size; (void)ws_size;

  float* votes = ws;
  float* awin  = votes + N_VOTES;
  float* Rw    = awin + N_AWIN;
  float* apb   = Rw + N_RW;
  float* S0    = apb + N_RW;
  float* S1    = S0 + N_S0;
  float* S2    = S1 + N_S1;
  float* mu    = S2 + N_S1;
  float* sigma = mu + N_S1;
  float* acts  = sigma + N_S1;
  float* rspm  = acts + N_S0;
  float* s1b   = rspm + N_S0;

  kInit<<<(N_RW + 255) / 256, 256, 0, stream>>>(x, awin, Rw);
  kVotes<<<57600 / 4, 128, 0, stream>>>(x, W, votes);
  for (int it = 0; it < 3; ++it) {
    kSpass<<<(N_S1 + 255) / 256, 256, 0, stream>>>(votes, Rw, awin, S0, S1, S2);
    kStats<<<(N_S0 + 255) / 256, 256, 0, stream>>>(S0, S1, S2, beta_v, lambda_, beta_a,
                                                   mu, sigma, acts, rspm, out);
    kAp<<<BC * 225, 128, 0, stream>>>(votes, sigma, acts, rspm, apb, s1b);
    kUpdate<<<(N_RW + 255) / 256, 256, 0, stream>>>(apb, s1b, Rw);
  }
  (void)mu;
}